// MyNet_55946243998332
// MI455X (gfx1250) — compile-verified
//
#include <hip/hip_runtime.h>

typedef float v2f __attribute__((ext_vector_type(2)));
typedef float v8f __attribute__((ext_vector_type(8)));

#define IN_F 1433
#define HID  16
#define NCLS 7

// ---------------- degree / normalization ----------------
__global__ void deg_init_k(float* deg, int n) {
    int i = blockIdx.x * blockDim.x + threadIdx.x;
    if (i < n) deg[i] = 1.0f;                      // self-loop contributes 1
}

__global__ void deg_count_k(const int* __restrict__ ei, float* deg, int nE) {
    int e = blockIdx.x * blockDim.x + threadIdx.x;
    if (e < nE) atomicAdd(&deg[ei[nE + e]], 1.0f); // dst = ei[1][e]
}

__global__ void dinv_k(float* deg, int n) {
    int i = blockIdx.x * blockDim.x + threadIdx.x;
    if (i < n) deg[i] = rsqrtf(deg[i]);            // deg >= 1 always
}

// ---------------- layer 1 GEMM: H1 = X @ W1 via V_WMMA_F32_16X16X4_F32 ----
// Wave handles a 16-row tile of X; N=16 covers all of HID.
// A layout (32-bit 16x4): lane<16 -> {K=0,K=1}, lane>=16 -> {K=2,K=3}; M=lane&15.
// B layout (4x16): v0 = row k (lanes 0-15) / row k+2 (lanes 16-31); v1 = k+1 / k+3.
__global__ __launch_bounds__(256) void gemm1_wmma_k(
        const float* __restrict__ X, const float* __restrict__ W1,
        float* __restrict__ H1, int nNodes, int nWaves) {
    int wave = blockIdx.x * (blockDim.x >> 5) + (threadIdx.x >> 5);
    if (wave >= nWaves) return;                    // wave-uniform: EXEC stays all-ones
    int lane = threadIdx.x & 31;
    int half = lane >> 4;                          // which K-pair this lane holds
    int n    = lane & 15;                          // N column / A row (M)
    int base = wave * 16;
    int arow = base + n;
    if (arow >= nNodes) arow = nNodes - 1;         // clamp (stores guarded below)
    const float* xrow = X + (long)arow * IN_F;

    v8f c = {0.f, 0.f, 0.f, 0.f, 0.f, 0.f, 0.f, 0.f};
    int k = 0;
    for (; k + 4 <= IN_F; k += 4) {
        v2f a, b;
        a.x = xrow[k + 2 * half];
        a.y = xrow[k + 2 * half + 1];
        b.x = W1[(k + 2 * half) * HID + n];
        b.y = W1[(k + 2 * half + 1) * HID + n];
        c = __builtin_amdgcn_wmma_f32_16x16x4_f32(
                /*neg_a=*/false, a, /*neg_b=*/false, b,
                /*c_mod=*/(short)0, c, /*reuse_a=*/false, /*reuse_b=*/false);
    }
    // K tail (1433 % 4 == 1): scalar FMA directly into the C fragment.
    for (; k < IN_F; ++k) {
        float wkn = W1[k * HID + n];
        #pragma unroll
        for (int j = 0; j < 8; ++j) {
            int row = base + j + 8 * half;
            int r = row < nNodes ? row : nNodes - 1;
            c[j] += X[(long)r * IN_F + k] * wkn;
        }
    }
    // C/D layout: vgpr j, lane -> element [M = j + 8*half][N = n]
    #pragma unroll
    for (int j = 0; j < 8; ++j) {
        int row = base + j + 8 * half;
        if (row < nNodes) H1[row * HID + n] = c[j];
    }
}

// ---------------- layer 1 aggregation ----------------
// A1[i,f] = b1[f] + H1[i,f]*dinv[i]^2  (self-loop term, also initializes buffer)
__global__ void selfloop1_k(const float* __restrict__ H1, const float* __restrict__ dinv,
                            const float* __restrict__ b1, float* A1, int n) {
    int t = blockIdx.x * blockDim.x + threadIdx.x;
    if (t >= n * HID) return;
    int i = t >> 4, f = t & 15;
    float dn = dinv[i];
    A1[t] = b1[f] + H1[t] * dn * dn;
}

__global__ void scatter1_k(const int* __restrict__ ei, const float* __restrict__ H1,
                           const float* __restrict__ dinv, float* A1, int nE) {
    int t = blockIdx.x * blockDim.x + threadIdx.x;
    if (t >= nE * HID) return;
    int e = t >> 4, f = t & 15;
    int s = ei[e], d = ei[nE + e];
    atomicAdd(&A1[d * HID + f], H1[s * HID + f] * dinv[s] * dinv[d]);
}

// ---------------- layer 2 GEMM (tiny): H2 = relu(A1) @ W2 ----------------
__global__ void gemm2_k(const float* __restrict__ A1, const float* __restrict__ W2,
                        float* __restrict__ H2, int n) {
    int i = blockIdx.x * blockDim.x + threadIdx.x;
    if (i >= n) return;
    float h[HID];
    #pragma unroll
    for (int f = 0; f < HID; ++f) h[f] = fmaxf(A1[i * HID + f], 0.f);
    #pragma unroll
    for (int cc = 0; cc < NCLS; ++cc) {
        float acc = 0.f;
        #pragma unroll
        for (int f = 0; f < HID; ++f) acc += h[f] * W2[f * NCLS + cc];
        H2[i * 8 + cc] = acc;                      // stride-8 rows for alignment
    }
}

// ---------------- layer 2 aggregation into d_out ----------------
__global__ void selfloop2_k(const float* __restrict__ H2, const float* __restrict__ dinv,
                            const float* __restrict__ b2, float* out, int n) {
    int i = blockIdx.x * blockDim.x + threadIdx.x;
    if (i >= n) return;
    float dn = dinv[i], dn2 = dn * dn;
    #pragma unroll
    for (int cc = 0; cc < NCLS; ++cc) out[i * NCLS + cc] = b2[cc] + H2[i * 8 + cc] * dn2;
}

__global__ void scatter2_k(const int* __restrict__ ei, const float* __restrict__ H2,
                           const float* __restrict__ dinv, float* out, int nE) {
    int t = blockIdx.x * blockDim.x + threadIdx.x;
    if (t >= nE * 8) return;
    int e = t >> 3, f = t & 7;
    if (f >= NCLS) return;
    int s = ei[e], d = ei[nE + e];
    atomicAdd(&out[d * NCLS + f], H2[s * 8 + f] * dinv[s] * dinv[d]);
}

extern "C" void kernel_launch(void* const* d_in, const int* in_sizes, int n_in,
                              void* d_out, int out_size, void* d_ws, size_t ws_size,
                              hipStream_t stream) {
    const float* X  = (const float*)d_in[0];
    const int*   ei = (const int*)  d_in[1];
    const float* W1 = (const float*)d_in[2];
    const float* b1 = (const float*)d_in[3];
    const float* W2 = (const float*)d_in[4];
    const float* b2 = (const float*)d_in[5];
    float* out = (float*)d_out;

    int nNodes = in_sizes[0] / IN_F;
    int nE     = in_sizes[1] / 2;

    // workspace layout (floats)
    float* dinv = (float*)d_ws;                    // nNodes
    float* H1   = dinv + ((nNodes + 63) & ~63);    // nNodes*16
    float* A1   = H1   + (size_t)nNodes * HID;     // nNodes*16
    float* H2   = A1   + (size_t)nNodes * HID;     // nNodes*8

    const int T = 256;
    deg_init_k <<<(nNodes + T - 1) / T, T, 0, stream>>>(dinv, nNodes);
    deg_count_k<<<(nE + T - 1) / T, T, 0, stream>>>(ei, dinv, nE);
    dinv_k     <<<(nNodes + T - 1) / T, T, 0, stream>>>(dinv, nNodes);

    int nWaves = (nNodes + 15) / 16;
    gemm1_wmma_k<<<(nWaves + 7) / 8, 256, 0, stream>>>(X, W1, H1, nNodes, nWaves);

    selfloop1_k<<<(nNodes * HID + T - 1) / T, T, 0, stream>>>(H1, dinv, b1, A1, nNodes);
    scatter1_k <<<(nE * HID + T - 1) / T, T, 0, stream>>>(ei, H1, dinv, A1, nE);

    gemm2_k    <<<(nNodes + T - 1) / T, T, 0, stream>>>(A1, W2, H2, nNodes);
    selfloop2_k<<<(nNodes + T - 1) / T, T, 0, stream>>>(H2, dinv, b2, out, nNodes);
    scatter2_k <<<(nE * 8 + T - 1) / T, T, 0, stream>>>(ei, H2, dinv, out, nE);
}